// HybridMdl_17540646437241
// MI455X (gfx1250) — compile-verified
//
#include <hip/hip_runtime.h>
#include <hip/hip_bf16.h>

typedef __attribute__((ext_vector_type(2))) float v2f;
typedef __attribute__((ext_vector_type(8))) float v8f;

namespace {

constexpr float kT1   = 0.01f;   // integration end time
constexpr float kDT   = 0.01f;   // initial step
constexpr float kRTOL = 1e-7f;
constexpr float kATOL = 1e-9f;

struct Ctrl { float t; float h; unsigned accept; unsigned cur; };

__device__ __forceinline__ float tanh_fast(float x) {
#if __has_builtin(__builtin_amdgcn_tanhf)
  return __builtin_amdgcn_tanhf(x);                      // v_tanh_f32 (gfx1250)
#else
  float e = __builtin_amdgcn_exp2f(x * 2.8853900817779268f);  // 2*log2(e)
  return 1.0f - 2.0f * __builtin_amdgcn_rcpf(e + 1.0f);
#endif
}

// Cross-half lane swap via ds_swizzle immediate (SWAPX16: xor=0x10, and=0x1f).
// Immediate-pattern swizzle: no address VGPR, no index-setup VALU (vs ds_bpermute).
__device__ __forceinline__ float swap16(float x) {
  return __int_as_float(__builtin_amdgcn_ds_swizzle(__float_as_int(x), 0x401F));
}
__device__ __forceinline__ float sw_xor(float x, int m) {  // group-of-32 xor swizzle
  switch (m) {
    case 1:  return __int_as_float(__builtin_amdgcn_ds_swizzle(__float_as_int(x), 0x041F));
    case 2:  return __int_as_float(__builtin_amdgcn_ds_swizzle(__float_as_int(x), 0x081F));
    case 4:  return __int_as_float(__builtin_amdgcn_ds_swizzle(__float_as_int(x), 0x101F));
    case 8:  return __int_as_float(__builtin_amdgcn_ds_swizzle(__float_as_int(x), 0x201F));
    default: return __int_as_float(__builtin_amdgcn_ds_swizzle(__float_as_int(x), 0x401F));
  }
}

#if __has_builtin(__builtin_amdgcn_wmma_f32_16x16x4_f32)
#define HAVE_WMMA4 1
#else
#define HAVE_WMMA4 0
#pragma message("__builtin_amdgcn_wmma_f32_16x16x4_f32 unavailable: scalar MLP fallback in use")
#endif

// Per-thread resident network parameters. With WMMA, the three 3x3 weight
// matrices are pre-marshaled into V_WMMA_F32_16X16X4_F32 A-operand layout:
// lane L(0-15): row M=L, VGPR0=K0, VGPR1=K1 ; lane L(16-31): row M=L-16, K2,K3.
struct Net {
#if HAVE_WMMA4
  v2f aW1, aW2, aWo;
  v8f cB1, cB2, cBo;  // C operands: rows 0-2 = bias broadcast (loop-invariant)
  bool hi;            // lane >= 16
#else
  float w1[9], w2[9], wo[9];
  float b1v[3], b2v[3];
#endif
  float bo0, bo1, bo2;
};

__device__ __forceinline__ Net make_net(const float* __restrict__ W1, const float* __restrict__ b1,
                                        const float* __restrict__ W2, const float* __restrict__ b2,
                                        const float* __restrict__ Wo, const float* __restrict__ bo) {
  Net nt;
  nt.bo0 = bo[0]; nt.bo1 = bo[1]; nt.bo2 = bo[2];
#if HAVE_WMMA4
  const int lane = threadIdx.x & 31;
  nt.hi = lane >= 16;
  const int row = lane & 15;
  const int kb  = (lane >> 4) * 2;                 // 0 or 2
  // Branchless, EXEC-uniform zero-padding of the 3x3 weights to 16x4.
  const float m0 = (row < 3 && kb     < 3) ? 1.f : 0.f;
  const float m1 = (row < 3 && kb + 1 < 3) ? 1.f : 0.f;
  const int rr = (row < 3 ? row : 0) * 3;
  const int c0 = (kb     < 3 ? kb     : 0);
  const int c1 = (kb + 1 < 3 ? kb + 1 : 0);
  nt.aW1.x = W1[rr + c0] * m0;  nt.aW1.y = W1[rr + c1] * m1;
  nt.aW2.x = W2[rr + c0] * m0;  nt.aW2.y = W2[rr + c1] * m1;
  nt.aWo.x = Wo[rr + c0] * m0;  nt.aWo.y = Wo[rr + c1] * m1;
  nt.cB1 = (v8f){b1[0], b1[1], b1[2], 0.f, 0.f, 0.f, 0.f, 0.f};
  nt.cB2 = (v8f){b2[0], b2[1], b2[2], 0.f, 0.f, 0.f, 0.f, 0.f};
  nt.cBo = (v8f){bo[0], bo[1], bo[2], 0.f, 0.f, 0.f, 0.f, 0.f};
#else
  #pragma unroll
  for (int i = 0; i < 9; ++i) { nt.w1[i] = W1[i]; nt.w2[i] = W2[i]; nt.wo[i] = Wo[i]; }
  #pragma unroll
  for (int i = 0; i < 3; ++i) { nt.b1v[i] = b1[i]; nt.b2v[i] = b2[i]; }
#endif
  return nt;
}

#if HAVE_WMMA4
__device__ __forceinline__ v8f wmma4(v2f a, v2f b, v8f c) {
  return __builtin_amdgcn_wmma_f32_16x16x4_f32(false, a, false, b, (short)0, c, false, false);
}
#endif

// f(y) = Lorenz terms + 2-layer tanh MLP + linear head, for 32 samples (lane==sample).
__device__ __forceinline__ void feval(const Net& nt, float x0, float x1, float x2,
                                      float& o0, float& o1, float& o2) {
  float g0 = 10.f * (x1 - x0);
  float g1 = x0 * (28.f - x2) - x1;
  float g2 = x0 * x1;
#if HAVE_WMMA4
  const bool hi = nt.hi;
  // ---- B operand marshal: yT (4x16) per 16-sample half ----
  float s0 = swap16(x0);
  float s1 = swap16(x1);
  float s2 = swap16(x2);
  v2f bA; bA.x = hi ? s2  : x0;  bA.y = hi ? 0.f : x1;   // samples 0-15
  v2f bB; bB.x = hi ? x2  : s0;  bB.y = hi ? 0.f : s1;   // samples 16-31
  // ---- layer 1: D rows 0-2 (lanes 0-15) = W1*yT + b1 ----
  v8f dA = wmma4(nt.aW1, bA, nt.cB1);
  v8f dB = wmma4(nt.aW1, bB, nt.cB1);
  float zA0 = tanh_fast(dA[0]), zA1 = tanh_fast(dA[1]), zA2 = tanh_fast(dA[2]);
  float zB0 = tanh_fast(dB[0]), zB1 = tanh_fast(dB[1]), zB2 = tanh_fast(dB[2]);
  // ---- layer 2 ----
  float uA = swap16(zA2);
  float uB = swap16(zB2);
  v2f b2A; b2A.x = hi ? uA : zA0;  b2A.y = hi ? 0.f : zA1;
  v2f b2B; b2B.x = hi ? uB : zB0;  b2B.y = hi ? 0.f : zB1;
  dA = wmma4(nt.aW2, b2A, nt.cB2);
  dB = wmma4(nt.aW2, b2B, nt.cB2);
  zA0 = tanh_fast(dA[0]); zA1 = tanh_fast(dA[1]); zA2 = tanh_fast(dA[2]);
  zB0 = tanh_fast(dB[0]); zB1 = tanh_fast(dB[1]); zB2 = tanh_fast(dB[2]);
  // ---- layer 3: C operand carries bout only (loop-invariant) ----
  uA = swap16(zA2);
  uB = swap16(zB2);
  v2f b3A; b3A.x = hi ? uA : zA0;  b3A.y = hi ? 0.f : zA1;
  v2f b3B; b3B.x = hi ? uB : zB0;  b3B.y = hi ? 0.f : zB1;
  dA = wmma4(nt.aWo, b3A, nt.cBo);
  dB = wmma4(nt.aWo, b3B, nt.cBo);
  // ---- merge halves back to lane==sample, add Lorenz residual per lane ----
  float m0 = swap16(dB[0]);
  float m1 = swap16(dB[1]);
  float m2 = swap16(dB[2]);
  o0 = (hi ? m0 : dA[0]) + g0;
  o1 = (hi ? m1 : dA[1]) + g1;
  o2 = (hi ? m2 : dA[2]) + g2;
#else
  float h0 = tanh_fast(nt.w1[0]*x0 + nt.w1[1]*x1 + nt.w1[2]*x2 + nt.b1v[0]);
  float h1 = tanh_fast(nt.w1[3]*x0 + nt.w1[4]*x1 + nt.w1[5]*x2 + nt.b1v[1]);
  float h2 = tanh_fast(nt.w1[6]*x0 + nt.w1[7]*x1 + nt.w1[8]*x2 + nt.b1v[2]);
  float p0 = tanh_fast(nt.w2[0]*h0 + nt.w2[1]*h1 + nt.w2[2]*h2 + nt.b2v[0]);
  float p1 = tanh_fast(nt.w2[3]*h0 + nt.w2[4]*h1 + nt.w2[5]*h2 + nt.b2v[1]);
  float p2 = tanh_fast(nt.w2[6]*h0 + nt.w2[7]*h1 + nt.w2[8]*h2 + nt.b2v[2]);
  o0 = g0 + nt.wo[0]*p0 + nt.wo[1]*p1 + nt.wo[2]*p2 + nt.bo0;
  o1 = g1 + nt.wo[3]*p0 + nt.wo[4]*p1 + nt.wo[5]*p2 + nt.bo1;
  o2 = g2 + nt.wo[6]*p0 + nt.wo[7]*p1 + nt.wo[8]*p2 + nt.bo2;
#endif
}

// ---------------- one dopri5 attempt for the whole batch ----------------
// Ping-pong: reads buf[cur], writes candidate into buf[cur^1].
__global__ __launch_bounds__(256) void step_kernel(
    float* __restrict__ buf0, float* __restrict__ buf1,
    const float* __restrict__ W1, const float* __restrict__ b1,
    const float* __restrict__ W2, const float* __restrict__ b2,
    const float* __restrict__ Wo, const float* __restrict__ bo,
    const Ctrl* __restrict__ ctrl, float* __restrict__ partials, int n) {
  const float t = ctrl->t;                     // uniform scalar load
  const float h = ctrl->h;
  if (t >= kT1 - 1e-12f) return;               // uniform early-out once integrated
  const float he = fminf(h, kT1 - t);
  const unsigned cur = ctrl->cur;              // uniform buffer parity
  const float* y_cur = cur ? buf1 : buf0;
  float*       y_new = cur ? buf0 : buf1;

  const int gid = blockIdx.x * 256 + threadIdx.x;
  const int i   = gid < n ? gid : (n - 1);     // clamp: keep EXEC full for WMMA
  const Net nt  = make_net(W1, b1, W2, b2, Wo, bo);

  float y0 = y_cur[3 * i], y1v = y_cur[3 * i + 1], y2v = y_cur[3 * i + 2];

  constexpr double AD[6][6] = {
    {0.2, 0, 0, 0, 0, 0},
    {3.0/40.0, 9.0/40.0, 0, 0, 0, 0},
    {44.0/45.0, -56.0/15.0, 32.0/9.0, 0, 0, 0},
    {19372.0/6561.0, -25360.0/2187.0, 64448.0/6561.0, -212.0/729.0, 0, 0},
    {9017.0/3168.0, -355.0/33.0, 46732.0/5247.0, 49.0/176.0, -5103.0/18656.0, 0},
    {35.0/384.0, 0.0, 500.0/1113.0, 125.0/192.0, -2187.0/6784.0, 11.0/84.0},
  };
  constexpr double BD[7]  = {35.0/384.0, 0.0, 500.0/1113.0, 125.0/192.0,
                             -2187.0/6784.0, 11.0/84.0, 0.0};
  constexpr double B4D[7] = {5179.0/57600.0, 0.0, 7571.0/16695.0, 393.0/640.0,
                             -92097.0/339200.0, 187.0/2100.0, 1.0/40.0};

  float k[7][3];
  feval(nt, y0, y1v, y2v, k[0][0], k[0][1], k[0][2]);   // stage 1 (FSAL-equal recompute)
  #pragma unroll
  for (int s = 1; s < 7; ++s) {
    float a0 = 0.f, a1 = 0.f, a2 = 0.f;
    #pragma unroll
    for (int j = 0; j < s; ++j) {
      if (AD[s - 1][j] != 0.0) {
        const float c = (float)AD[s - 1][j];
        a0 += c * k[j][0]; a1 += c * k[j][1]; a2 += c * k[j][2];
      }
    }
    feval(nt, y0 + he * a0, y1v + he * a1, y2v + he * a2,
          k[s][0], k[s][1], k[s][2]);
  }

  float sy0 = 0.f, sy1 = 0.f, sy2 = 0.f, se0 = 0.f, se1 = 0.f, se2 = 0.f;
  #pragma unroll
  for (int j = 0; j < 7; ++j) {
    if (BD[j] != 0.0) {
      const float c = (float)BD[j];
      sy0 += c * k[j][0]; sy1 += c * k[j][1]; sy2 += c * k[j][2];
    }
    if (BD[j] - B4D[j] != 0.0) {
      const float c = (float)(BD[j] - B4D[j]);
      se0 += c * k[j][0]; se1 += c * k[j][1]; se2 += c * k[j][2];
    }
  }
  const float n0 = y0  + he * sy0;
  const float n1 = y1v + he * sy1;
  const float n2 = y2v + he * sy2;
  const float e0 = he * se0, e1 = he * se1, e2 = he * se2;
  const float tol0 = kATOL + kRTOL * fmaxf(fabsf(y0),  fabsf(n0));
  const float tol1 = kATOL + kRTOL * fmaxf(fabsf(y1v), fabsf(n1));
  const float tol2 = kATOL + kRTOL * fmaxf(fabsf(y2v), fabsf(n2));
  const float r0 = e0 * __builtin_amdgcn_rcpf(tol0);
  const float r1 = e1 * __builtin_amdgcn_rcpf(tol1);
  const float r2 = e2 * __builtin_amdgcn_rcpf(tol2);
  float v = r0 * r0 + r1 * r1 + r2 * r2;
  v = (gid < n) ? v : 0.f;                     // clamped duplicates must not count

  y_new[3 * i] = n0; y_new[3 * i + 1] = n1; y_new[3 * i + 2] = n2;

  // Deterministic block partial: fixed-order wave butterfly + fixed-order merge.
  v += sw_xor(v, 1);  v += sw_xor(v, 2);
  v += sw_xor(v, 4);  v += sw_xor(v, 8);
  v += sw_xor(v, 16);
  __shared__ float red[8];
  if ((threadIdx.x & 31) == 0) red[threadIdx.x >> 5] = v;
  __syncthreads();
  if (threadIdx.x == 0) {
    partials[blockIdx.x] = ((red[0] + red[1]) + (red[2] + red[3])) +
                           ((red[4] + red[5]) + (red[6] + red[7]));
  }
}

// ------------- global controller: enorm, accept, t/h/parity update -------------
__global__ __launch_bounds__(256) void update_kernel(Ctrl* __restrict__ ctrl,
                                                     const float* __restrict__ partials,
                                                     int nblk, int n) {
  __shared__ float s[256];
  const float t = ctrl->t;
  const float h = ctrl->h;
  const bool done = t >= kT1 - 1e-12f;
  float acc = 0.f;
  if (!done) {
    for (int idx = threadIdx.x; idx < nblk; idx += 256) acc += partials[idx];  // fixed order
  }
  s[threadIdx.x] = acc;
  __syncthreads();
  #pragma unroll
  for (int off = 128; off > 0; off >>= 1) {
    if (threadIdx.x < off) s[threadIdx.x] += s[threadIdx.x + off];
    __syncthreads();
  }
  if (threadIdx.x == 0) {
    if (done) {
      ctrl->accept = 0u;
    } else {
      const float he = fminf(h, kT1 - t);
      const float enorm = __builtin_amdgcn_sqrtf(s[0] / (3.0f * (float)n));
      const bool accept = enorm <= 1.0f;
      const float em = fmaxf(enorm, 1e-10f);
      float factor = 0.9f * __builtin_amdgcn_exp2f(-0.2f * __builtin_amdgcn_logf(em));
      factor = fminf(fmaxf(factor, 0.2f), 10.f);
      ctrl->h = he * factor;
      if (accept) { ctrl->t = t + he; ctrl->cur ^= 1u; }  // commit by parity flip
      ctrl->accept = accept ? 1u : 0u;
    }
  }
}

// Final commit: state must end in d_out (== buf0). Copy only if parity is on buf1.
__global__ __launch_bounds__(256) void commit_kernel(float* __restrict__ buf0,
                                                     const float* __restrict__ buf1,
                                                     const Ctrl* __restrict__ ctrl, int total) {
  if (ctrl->cur == 0u) return;                 // uniform
  const int gid = blockIdx.x * 256 + threadIdx.x;
  if (gid < total) buf0[gid] = buf1[gid];
}

__global__ __launch_bounds__(256) void init_kernel(const float* __restrict__ inp,
                                                   float* __restrict__ y,
                                                   Ctrl* __restrict__ ctrl, int total) {
  const int gid = blockIdx.x * 256 + threadIdx.x;
  if (gid < total) y[gid] = inp[gid];
  if (gid == 0) { ctrl->t = 0.f; ctrl->h = kDT; ctrl->accept = 0u; ctrl->cur = 0u; }
}

} // namespace

extern "C" void kernel_launch(void* const* d_in, const int* in_sizes, int n_in,
                              void* d_out, int out_size, void* d_ws, size_t ws_size,
                              hipStream_t stream) {
  const float* inp  = (const float*)d_in[0];
  const float* W1   = (const float*)d_in[1];
  const float* b1   = (const float*)d_in[2];
  const float* W2   = (const float*)d_in[3];
  const float* b2   = (const float*)d_in[4];
  const float* Wout = (const float*)d_in[5];
  const float* bout = (const float*)d_in[6];
  float* out = (float*)d_out;

  const int n     = in_sizes[0] / 3;
  const int total = 3 * n;
  const int nblk  = (n + 255) / 256;
  const int cgrid = (total + 255) / 256;

  char* wsb = (char*)d_ws;
  Ctrl*  ctrl     = (Ctrl*)wsb;                                   // 16 B
  float* partials = (float*)(wsb + 256);                          // nblk floats
  size_t yoff     = (256 + (size_t)nblk * sizeof(float) + 255) & ~(size_t)255;
  float* y1buf    = (float*)(wsb + yoff);                         // ping-pong buffer, 12 MB

  init_kernel<<<cgrid, 256, 0, stream>>>(inp, out, ctrl, total);
  for (int it = 0; it < 16; ++it) {   // MAX_STEPS; each no-ops uniformly once t >= T1
    step_kernel<<<nblk, 256, 0, stream>>>(out, y1buf, W1, b1, W2, b2, Wout, bout,
                                          ctrl, partials, n);
    update_kernel<<<1, 256, 0, stream>>>(ctrl, partials, nblk, n);
  }
  commit_kernel<<<cgrid, 256, 0, stream>>>(out, y1buf, ctrl, total);
}